// HLCLConv_72559177498822
// MI455X (gfx1250) — compile-verified
//
#include <hip/hip_runtime.h>

// ---------------------------------------------------------------------------
// 2-layer GCN (HLCLConv low-pass) for MI455X / gfx1250.
// Heavy cost is the edge scatter (~1.7 GB HBM traffic -> ~73us at 23.3 TB/s);
// GEMMs (2x 1.64 GFLOP) done with V_WMMA_F32_16X16X4_F32 at full f32 precision.
// ---------------------------------------------------------------------------

typedef __attribute__((ext_vector_type(2))) float v2f;
typedef __attribute__((ext_vector_type(8))) float v8f;

#define GCN_N 50000
#define GCN_E 800000
#define GCN_H 128

// ---------------- tiny utility kernels ----------------

__global__ __launch_bounds__(256) void k_fill(float* __restrict__ p, float v, int n) {
  int i = blockIdx.x * 256 + threadIdx.x;
  if (i < n) p[i] = v;
}

__global__ __launch_bounds__(256) void k_zero4(float4* __restrict__ p, int n4) {
  int i = blockIdx.x * 256 + threadIdx.x;
  if (i < n4) p[i] = make_float4(0.f, 0.f, 0.f, 0.f);
}

// deg[dst] += w  (self-loop "+1" pre-seeded by k_fill(deg, 1.0f))
__global__ __launch_bounds__(256) void k_deg_accum(const int* __restrict__ dst,
                                                   const float* __restrict__ w,
                                                   float* __restrict__ deg, int n) {
  int e = blockIdx.x * 256 + threadIdx.x;
  if (e < n) unsafeAtomicAdd(&deg[dst[e]], w[e]);
}

// deg -> deg^{-1/2} in place
__global__ __launch_bounds__(256) void k_inv_sqrt(float* __restrict__ deg, int n) {
  int i = blockIdx.x * 256 + threadIdx.x;
  if (i < n) {
    float d = deg[i];
    deg[i] = (d > 0.f) ? rsqrtf(fmaxf(d, 1e-38f)) : 0.f;
  }
}

// norm[e] = dis[src]*w*dis[dst]
__global__ __launch_bounds__(256) void k_norm(const int* __restrict__ src,
                                              const int* __restrict__ dst,
                                              const float* __restrict__ w,
                                              const float* __restrict__ dis,
                                              float* __restrict__ norm, int n) {
  int e = blockIdx.x * 256 + threadIdx.x;
  if (e < n) norm[e] = dis[src[e]] * w[e] * dis[dst[e]];
}

// ---------------- WMMA f32 GEMM: C[M,128] = A[M,128] @ B[128,128] ----------------
// One wave (32 lanes) per 16-row band; 8 accumulators cover all 128 output cols,
// so the A fragment is loaded once per K-step and reused by 8 WMMAs.
// ISA VGPR layouts (CDNA5 ISA):
//   A 16x4 f32 : lane m=lane%16, v2f = {A[m][kb], A[m][kb+1]}, kb = 2*(lane/16)
//   B 4x16 f32 : lane n=lane%16, v2f = {B[kb][n],  B[kb+1][n]}
//   C/D 16x16  : 8 VGPRs, element [v] = C[v + 8*(lane/16)][lane%16]
__global__ __launch_bounds__(32) void k_gemm_wmma(const float* __restrict__ A,
                                                  const float* __restrict__ B,
                                                  float* __restrict__ C) {
  const int tile = blockIdx.x;          // 16-row band
  const int lane = threadIdx.x & 31;
  const int half = lane >> 4;           // 0 or 1
  const int l16  = lane & 15;
  const int row  = tile * 16 + l16;
  const int kb   = half * 2;            // K sub-offset: 0 or 2

  v8f acc[8];
#pragma unroll
  for (int j = 0; j < 8; ++j) acc[j] = (v8f){0.f,0.f,0.f,0.f,0.f,0.f,0.f,0.f};

  for (int k0 = 0; k0 < GCN_H; k0 += 4) {
    // A fragment: 8-byte aligned vector load
    v2f a = *(const v2f*)(A + (size_t)row * GCN_H + k0 + kb);
#pragma unroll
    for (int j = 0; j < 8; ++j) {
      int n = j * 16 + l16;
      v2f b;
      b.x = B[(size_t)(k0 + kb) * GCN_H + n];
      b.y = B[(size_t)(k0 + kb + 1) * GCN_H + n];
      acc[j] = __builtin_amdgcn_wmma_f32_16x16x4_f32(
          /*neg_a=*/false, a, /*neg_b=*/false, b,
          /*c_mod=*/(short)0, acc[j], /*reuse_a=*/false, /*reuse_b=*/false);
    }
  }

  const int mbase = tile * 16 + half * 8;
#pragma unroll
  for (int j = 0; j < 8; ++j) {
#pragma unroll
    for (int v = 0; v < 8; ++v) {
      C[(size_t)(mbase + v) * GCN_H + j * 16 + l16] = acc[j][v];
    }
  }
}

// ---------------- edge scatter: agg[dst] += norm_e * zw[src] ----------------
// One wave per edge, float4 per lane (32*4 = 128 features), hw f32 atomics.
__global__ __launch_bounds__(256) void k_scatter(const float* __restrict__ zw,
                                                 const float* __restrict__ norm,
                                                 const int* __restrict__ src,
                                                 const int* __restrict__ dst,
                                                 float* __restrict__ agg, int nEdges) {
  int e = blockIdx.x * 8 + (threadIdx.x >> 5);
  if (e >= nEdges) return;
  int lane = threadIdx.x & 31;
  int s = src[e], d = dst[e];
  float w = norm[e];
  float4 v = *((const float4*)(zw + (size_t)s * GCN_H) + lane);
  float* op = agg + (size_t)d * GCN_H + lane * 4;
  unsafeAtomicAdd(op + 0, w * v.x);
  unsafeAtomicAdd(op + 1, w * v.y);
  unsafeAtomicAdd(op + 2, w * v.z);
  unsafeAtomicAdd(op + 3, w * v.w);
}

// ---------------- finalize: out = relu(agg + dis[i]^2 * zw + bias) ----------------
// (self-loop contribution norm_ii = dis[i]*1*dis[i] folded in here)
__global__ __launch_bounds__(256) void k_finalize(const float* __restrict__ agg,
                                                  const float* __restrict__ zw,
                                                  const float* __restrict__ dis,
                                                  const float* __restrict__ bias,
                                                  float* __restrict__ out, int nElem) {
  int i = blockIdx.x * 256 + threadIdx.x;
  if (i >= nElem) return;
  int node = i >> 7;      // /128
  int c    = i & 127;
  float dn = dis[node];
  float v = agg[i] + dn * dn * zw[i] + bias[c];
  out[i] = v > 0.f ? v : 0.f;
}

// ---------------- host-side launch ----------------

extern "C" void kernel_launch(void* const* d_in, const int* in_sizes, int n_in,
                              void* d_out, int out_size, void* d_ws, size_t ws_size,
                              hipStream_t stream) {
  const float* x  = (const float*)d_in[0];   // [N,128]
  const int*   ei = (const int*)d_in[1];     // [2,E]
  const float* ew = (const float*)d_in[2];   // [E]
  const float* W1 = (const float*)d_in[3];   // [128,128]
  const float* b1 = (const float*)d_in[4];   // [128]
  const float* W2 = (const float*)d_in[5];   // [128,128]
  const float* b2 = (const float*)d_in[6];   // [128]
  float* out = (float*)d_out;                // [N,128]

  const int N = GCN_N, E = GCN_E, H = GCN_H;
  const int* src = ei;
  const int* dst = ei + E;

  // workspace layout (floats): dis[N] | norm[E] | zw[N*H] | agg[N*H]  (~54.6 MB)
  float* ws   = (float*)d_ws;
  float* dis  = ws;
  float* norm = ws + N;
  float* zw   = ws + N + E;
  float* agg  = zw + (size_t)N * H;

  const int NH   = N * H;        // 6,400,000
  const int NH4  = NH / 4;
  const int gN   = (N + 255) / 256;
  const int gE   = (E + 255) / 256;
  const int gNH  = (NH + 255) / 256;
  const int gNH4 = (NH4 + 255) / 256;
  const int gTiles = N / 16;     // 3125 (exact)
  const int gScat  = (E + 7) / 8;

  // --- normalization (shared by both layers) ---
  k_fill<<<gN, 256, 0, stream>>>(dis, 1.0f, N);                       // self-loop weight
  k_deg_accum<<<gE, 256, 0, stream>>>(dst, ew, dis, E);               // deg
  k_inv_sqrt<<<gN, 256, 0, stream>>>(dis, N);                         // deg^-1/2
  k_norm<<<gE, 256, 0, stream>>>(src, dst, ew, dis, norm, E);

  // --- layer 1 ---
  k_gemm_wmma<<<gTiles, 32, 0, stream>>>(x, W1, zw);                  // zw = x @ W1
  k_zero4<<<gNH4, 256, 0, stream>>>((float4*)agg, NH4);
  k_scatter<<<gScat, 256, 0, stream>>>(zw, norm, src, dst, agg, E);
  k_finalize<<<gNH, 256, 0, stream>>>(agg, zw, dis, b1, agg, NH);     // z1 in agg

  // --- layer 2 ---
  k_gemm_wmma<<<gTiles, 32, 0, stream>>>(agg, W2, zw);                // zw = z1 @ W2
  k_zero4<<<gNH4, 256, 0, stream>>>((float4*)agg, NH4);
  k_scatter<<<gScat, 256, 0, stream>>>(zw, norm, src, dst, agg, E);
  k_finalize<<<gNH, 256, 0, stream>>>(agg, zw, dis, b2, out, NH);
}